// FlowMatching_31044023615894
// MI455X (gfx1250) — compile-verified
//
#include <hip/hip_runtime.h>
#include <math.h>

typedef __attribute__((ext_vector_type(2))) float        v2f;
typedef __attribute__((ext_vector_type(8))) float        v8f;
typedef __attribute__((ext_vector_type(4))) unsigned int v4u;
typedef __attribute__((ext_vector_type(4))) int          v4i;
typedef __attribute__((ext_vector_type(8))) int          v8i;

#define B_  16
#define N_  1024
#define D_  8
#define C_  128
#define H_  256
#define NT_ (N_/16)   /* 64 n-tiles  */
#define HT_ (H_/16)   /* 16 h-tiles  */

static __device__ __forceinline__ v8f wmma4(v2f a, v2f b, v8f c) {
  // V_WMMA_F32_16X16X4_F32 : D = A(16x4,f32) * B(4x16,f32) + C(16x16,f32)
  return __builtin_amdgcn_wmma_f32_16x16x4_f32(false, a, false, b, (short)0, c,
                                               false, false);
}

// ---------------------------------------------------------------------------
// x0 init (cond dims <- data, free dims <- noise), ||data_m||^2, zero counters
// ---------------------------------------------------------------------------
__global__ void k_init(const float* __restrict__ data,
                       const float* __restrict__ x0n,
                       float* __restrict__ x0, float* __restrict__ ynorm,
                       int* __restrict__ counts) {
  int idx = blockIdx.x * blockDim.x + threadIdx.x;   // b*N + n
  if (idx < 8) counts[idx] = 0;
  if (idx >= B_ * N_) return;
  float s = 0.f;
#pragma unroll
  for (int d = 0; d < D_; ++d) {
    float dv = data[idx * D_ + d];
    x0[idx * D_ + d] = (d < 3) ? dv : x0n[idx * D_ + d];
    s += dv * dv;
  }
  ynorm[idx] = s;
}

// ---------------------------------------------------------------------------
// cwcf[b][h] = (c @ Wc)[b,h] + t[b]*Wt[h] + b1[h]   — one wave per H-tile,
// A = c (M=16 batches, K=128) -> 32 WMMA k-steps.
// ---------------------------------------------------------------------------
__global__ void k_cwc(const float* __restrict__ c, const float* __restrict__ Wc,
                      const float* __restrict__ t, const float* __restrict__ Wt,
                      const float* __restrict__ b1, float* __restrict__ cwcf) {
  int j = blockIdx.x;
  int lane = threadIdx.x, hi = lane >> 4, lr = lane & 15;
  v8f acc = {};
#pragma unroll 4
  for (int ks = 0; ks < C_ / 4; ++ks) {
    int kb = ks * 4 + 2 * hi;
    v2f a, b;
    a.x = c[lr * C_ + kb];            a.y = c[lr * C_ + kb + 1];
    b.x = Wc[kb * H_ + j * 16 + lr];  b.y = Wc[(kb + 1) * H_ + j * 16 + lr];
    acc = wmma4(a, b, acc);
  }
  int h = j * 16 + lr;
  float bias = b1[h], wt = Wt[h];
#pragma unroll
  for (int r = 0; r < 8; ++r) {
    int bb = r + 8 * hi;                         // C row = batch
    cwcf[bb * H_ + h] = acc[r] + t[bb] * wt + bias;
  }
}

// ---------------------------------------------------------------------------
// TDM descriptor helper: 1-D copy of `elems` f32 from `gsrc` to LDS `lds_off`,
// optional LDS padding of +1 DWORD per 8 DWORDs (pad rows 8->9 for bank spread)
// D# layout per CDNA5 ISA (group0 + group1; 2D-max form).
// ---------------------------------------------------------------------------
static __device__ __forceinline__ void tdm_load_1d(const float* gsrc,
                                                   unsigned lds_off,
                                                   unsigned elems, bool pad) {
  unsigned long long ga = (unsigned long long)(size_t)gsrc;
  v4u g0 = { 1u,                                  // count=1 (valid), no gather
             lds_off,                             // lds_addr (bytes)
             (unsigned)ga,                        // global_addr[31:0]
             (unsigned)((ga >> 32) & 0x1FFFFFFu) | (2u << 30) }; // addr[56:32] | type=2
  unsigned dw0 = (2u << 16)                       // data_size = 4 bytes
               | (pad ? (1u << 20) : 0u)          // pad_enable
               | (pad ? (2u << 22) : 0u)          // pad_interval: 8 DWORDs
               | 0u;                              // pad_amount: 1 DWORD (code 0)
  v8i g1 = { (int)dw0,
             (int)(elems << 16),                  // tensor_dim0[15:0] (<<16), bar_addr=0
             (int)((1u << 16) | (elems >> 16)),   // tensor_dim0[31:16]=0, tensor_dim1=1
             (int)(elems << 16),                  // tile_dim0 = elems
             (int)1,                              // tile_dim1 = 1, tile_dim2 = 0
             (int)elems,                          // tensor_dim0_stride[31:0]
             (int)(elems << 16),                  // tensor_dim1_stride[15:0]
             0 };
  v4i z4 = { 0, 0, 0, 0 };                        // groups 2/3 unused (<=2D tensor)
  v8i z8 = { 0, 0, 0, 0, 0, 0, 0, 0 };
  __builtin_amdgcn_tensor_load_to_lds(g0, g1, z4, z4, z8, 0);
}

// ---------------------------------------------------------------------------
// Per batch (one 32-wave workgroup): TDM-stage data[b] (padded 9-dword rows)
// + ynorm[b] into LDS, then each wave computes argmin over m for 2 n-tiles:
//   score(n,m) = ||y_m||^2 - 2 x_n . y_m  (sqrt/||x||^2 dropped: argmin-inv.)
// ---------------------------------------------------------------------------
__global__ void __launch_bounds__(1024)
k_mismatch(const float* __restrict__ x0, const float* __restrict__ data,
           const float* __restrict__ ynorm, int* __restrict__ mask,
           int* __restrict__ counts, int iter) {
  __shared__ float yds[N_ * 9];     // 1024 rows x (8 data + 1 pad) dwords = 36KB
  __shared__ float ynl[N_];         // 4KB
  int b = blockIdx.x;
  int tid = threadIdx.x;
  int w = tid >> 5, lane = tid & 31, hi = lane >> 4, lr = lane & 15;

  if (tid < 32) {                   // wave 0 issues both DMAs
    tdm_load_1d(data + (size_t)b * N_ * D_,
                (unsigned)(size_t)(void*)yds, N_ * D_, true);
    tdm_load_1d(ynorm + (size_t)b * N_,
                (unsigned)(size_t)(void*)ynl, N_, false);
    __builtin_amdgcn_s_wait_tensorcnt(0);
  }
  __syncthreads();

  const float* xb = x0 + (size_t)b * N_ * D_;
  int nt0 = w, nt1 = w + 32;
  int r0 = nt0 * 16 + lr, r1 = nt1 * 16 + lr;
  v2f a00, a01, a10, a11;           // A frags for both n-tiles, K=0..7
  a00.x = xb[r0 * D_ + 2 * hi];     a00.y = xb[r0 * D_ + 2 * hi + 1];
  a01.x = xb[r0 * D_ + 4 + 2 * hi]; a01.y = xb[r0 * D_ + 5 + 2 * hi];
  a10.x = xb[r1 * D_ + 2 * hi];     a10.y = xb[r1 * D_ + 2 * hi + 1];
  a11.x = xb[r1 * D_ + 4 + 2 * hi]; a11.y = xb[r1 * D_ + 5 + 2 * hi];

  float mv0[8], mv1[8]; int mi0[8], mi1[8];
#pragma unroll
  for (int r = 0; r < 8; ++r) {
    mv0[r] = 3.4e38f; mv1[r] = 3.4e38f; mi0[r] = N_; mi1[r] = N_;
  }

  for (int mt = 0; mt < NT_; ++mt) {
    int m = mt * 16 + lr;
    const float* yr = &yds[m * 9 + 2 * hi];   // padded row: stride 9 dwords
    v2f b0, b1v;                               // B frags shared by both tiles
    b0.x = yr[0];  b0.y = yr[1];
    b1v.x = yr[4]; b1v.y = yr[5];
    float gn = ynl[m];
    v8f acc0 = {}, acc1 = {};
    acc0 = wmma4(a00, b0, acc0);
    acc0 = wmma4(a01, b1v, acc0);
    acc1 = wmma4(a10, b0, acc1);
    acc1 = wmma4(a11, b1v, acc1);
#pragma unroll
    for (int r = 0; r < 8; ++r) {
      float s0 = gn - 2.f * acc0[r];
      if (s0 < mv0[r] || (s0 == mv0[r] && m < mi0[r])) { mv0[r] = s0; mi0[r] = m; }
      float s1 = gn - 2.f * acc1[r];
      if (s1 < mv1[r] || (s1 == mv1[r] && m < mi1[r])) { mv1[r] = s1; mi1[r] = m; }
    }
  }
  // butterfly min-reduce across the 16 lanes of each half
  for (int off = 1; off < 16; off <<= 1) {
#pragma unroll
    for (int r = 0; r < 8; ++r) {
      float ov = __shfl_xor(mv0[r], off, 16);
      int   oi = __shfl_xor(mi0[r], off, 16);
      if (ov < mv0[r] || (ov == mv0[r] && oi < mi0[r])) { mv0[r] = ov; mi0[r] = oi; }
      ov = __shfl_xor(mv1[r], off, 16);
      oi = __shfl_xor(mi1[r], off, 16);
      if (ov < mv1[r] || (ov == mv1[r] && oi < mi1[r])) { mv1[r] = ov; mi1[r] = oi; }
    }
  }
  if (lr == 0) {
    int mis = 0;
#pragma unroll
    for (int r = 0; r < 8; ++r) {
      int rg0 = nt0 * 16 + r + 8 * hi;
      int bad0 = (mi0[r] != rg0) ? 1 : 0;
      mask[b * N_ + rg0] = bad0;
      int rg1 = nt1 * 16 + r + 8 * hi;
      int bad1 = (mi1[r] != rg1) ? 1 : 0;
      mask[b * N_ + rg1] = bad1;
      mis += bad0 + bad1;
    }
    if (mis) atomicAdd(&counts[iter], mis);
  }
}

// ---------------------------------------------------------------------------
// Renoise step i of the while_loop: predicated on counts[i] >= 10.
// ---------------------------------------------------------------------------
__global__ void k_renoise(float* __restrict__ x0,
                          const float* __restrict__ nbk,
                          const int* __restrict__ mask,
                          const int* __restrict__ counts, int iter) {
  if (counts[iter] < 10) return;                 // while-loop exit condition
  int idx = blockIdx.x * blockDim.x + threadIdx.x;
  if (idx >= B_ * N_) return;
  if (!mask[idx]) return;
  const float* nb = nbk + ((size_t)iter * B_ * N_ + idx) * D_;
#pragma unroll
  for (int d = 3; d < D_; ++d) x0[idx * D_ + d] = nb[d];  // cond dims stay = data
}

// ---------------------------------------------------------------------------
// Fused model + loss partials per (b, n-tile):
//  xt -> z = xt@W1 (+bias) -> tanh -> vt = h@W2 (h staged via LDS) -> MSE part.
// ---------------------------------------------------------------------------
__global__ void k_model(const float* __restrict__ x0,
                        const float* __restrict__ data,
                        const float* __restrict__ t,
                        const float* __restrict__ W1,
                        const float* __restrict__ cwcf,
                        const float* __restrict__ W2,
                        float* __restrict__ lpart) {
  __shared__ float hsh[16][17];
  int b = blockIdx.y, nt = blockIdx.x;
  int lane = threadIdx.x, hi = lane >> 4, lr = lane & 15;
  float tb = t[b];
  int row = nt * 16 + lr;
  const float* xr = x0 + ((size_t)b * N_ + row) * D_;
  const float* dr = data + ((size_t)b * N_ + row) * D_;
  auto xt = [&](int d) -> float {
    float dv = dr[d];
    return (d < 3) ? dv : ((1.f - tb) * xr[d] + tb * dv);   // cond dims = data exactly
  };
  v2f a0, a1;
  a0.x = xt(2 * hi);     a0.y = xt(2 * hi + 1);
  a1.x = xt(4 + 2 * hi); a1.y = xt(5 + 2 * hi);

  v8f accv = {};                                 // vt tile (cols d<8 valid)
  for (int ht = 0; ht < HT_; ++ht) {
    int h0 = ht * 16;
    v2f b0, b1v;
    b0.x  = W1[(2 * hi) * H_ + h0 + lr];      b0.y  = W1[(2 * hi + 1) * H_ + h0 + lr];
    b1v.x = W1[(4 + 2 * hi) * H_ + h0 + lr];  b1v.y = W1[(5 + 2 * hi) * H_ + h0 + lr];
    v8f z = {};
    z = wmma4(a0, b0, z);
    z = wmma4(a1, b1v, z);
    float bias = cwcf[b * H_ + h0 + lr];
    __syncthreads();
#pragma unroll
    for (int r = 0; r < 8; ++r)
      hsh[r + 8 * hi][lr] = tanhf(z[r] + bias);  // [n-local][h-local]
    __syncthreads();
#pragma unroll
    for (int kk = 0; kk < 4; ++kk) {             // K=16 chunk of h @ W2
      v2f ah, bw;
      ah.x = hsh[lr][kk * 4 + 2 * hi];
      ah.y = hsh[lr][kk * 4 + 2 * hi + 1];
      int hk = h0 + kk * 4 + 2 * hi;
      bw.x = (lr < D_) ? W2[hk * D_ + lr] : 0.f;         // pad cols 8..15
      bw.y = (lr < D_) ? W2[(hk + 1) * D_ + lr] : 0.f;
      accv = wmma4(ah, bw, accv);
    }
  }
  // masked MSE: only free dims d=3..7 contribute
  float local = 0.f;
  if (lr >= 3 && lr < D_) {
#pragma unroll
    for (int r = 0; r < 8; ++r) {
      int rg = nt * 16 + r + 8 * hi;
      size_t o = ((size_t)b * N_ + rg) * D_ + lr;
      float ut = data[o] - x0[o];
      float df = accv[r] - ut;
      local += df * df;
    }
  }
  for (int off = 1; off < 32; off <<= 1) local += __shfl_xor(local, off, 32);
  if (lane == 0) lpart[b * NT_ + nt] = local;
}

// ---------------------------------------------------------------------------
// Deterministic final reduction: loss[b] = sum(partials)/(N*N_FREE)
// ---------------------------------------------------------------------------
__global__ void k_loss_final(const float* __restrict__ lpart,
                             float* __restrict__ out) {
  int b = threadIdx.x;
  if (b >= B_) return;
  float s = 0.f;
  for (int i = 0; i < NT_; ++i) s += lpart[b * NT_ + i];
  out[b] = s * (1.0f / (N_ * 5));                // N_FREE = 5
}

extern "C" void kernel_launch(void* const* d_in, const int* in_sizes, int n_in,
                              void* d_out, int out_size, void* d_ws, size_t ws_size,
                              hipStream_t stream) {
  const float* data = (const float*)d_in[0];
  const float* c    = (const float*)d_in[1];
  const float* t    = (const float*)d_in[2];
  const float* x0n  = (const float*)d_in[3];
  const float* nbk  = (const float*)d_in[4];
  const float* W1   = (const float*)d_in[5];
  const float* Wc   = (const float*)d_in[6];
  const float* Wt   = (const float*)d_in[7];
  const float* b1   = (const float*)d_in[8];
  const float* W2   = (const float*)d_in[9];
  float* out = (float*)d_out;

  // workspace layout (~680 KB)
  float* x0     = (float*)d_ws;            // B*N*D
  float* ynorm  = x0 + B_ * N_ * D_;       // B*N
  float* cwcf   = ynorm + B_ * N_;         // B*H
  float* lpart  = cwcf + B_ * H_;          // B*NT
  int*   mask   = (int*)(lpart + B_ * NT_);// B*N
  int*   counts = mask + B_ * N_;          // 8

  k_init<<<dim3((B_ * N_ + 255) / 256), dim3(256), 0, stream>>>(data, x0n, x0,
                                                                ynorm, counts);
  k_cwc<<<dim3(HT_), dim3(32), 0, stream>>>(c, Wc, t, Wt, b1, cwcf);
  for (int i = 0; i < 4; ++i) {            // unrolled lax.while_loop (max 4 iters)
    k_mismatch<<<dim3(B_), dim3(1024), 0, stream>>>(x0, data, ynorm, mask,
                                                    counts, i);
    k_renoise<<<dim3((B_ * N_ + 255) / 256), dim3(256), 0, stream>>>(x0, nbk,
                                                                     mask, counts, i);
  }
  k_model<<<dim3(NT_, B_), dim3(32), 0, stream>>>(x0, data, t, W1, cwcf, W2, lpart);
  k_loss_final<<<dim3(1), dim3(32), 0, stream>>>(lpart, out);
}